// SetCriterion_48876727829092
// MI455X (gfx1250) — compile-verified
//
#include <hip/hip_runtime.h>
#include <math.h>

// Problem sizes from setup_inputs(): B=64, Q=4096, G=256, C=128 (all powers of 2).
#define B_ 64
#define Q_ 4096
#define G_ 256
#define C_ 128
#define BQ_ (B_ * Q_)
#define N_LOGITS ((long long)B_ * Q_ * C_)
#define N4_ ((int)(N_LOGITS / 4))          // 8,388,608 float4 elements

#define FOCAL_BLOCKS 4096
#define FOCAL_THREADS 256
#define GEOM_THREADS 64
#define GEOM_BLOCKS (BQ_ / GEOM_THREADS)   // 4096
#define LAB_THREADS 256
#define LAB_BLOCKS (BQ_ / LAB_THREADS)     // 1024

typedef __attribute__((ext_vector_type(2))) float v2f;
typedef __attribute__((ext_vector_type(4))) float v4f;
typedef __attribute__((ext_vector_type(8))) float v8f;

// ---------------------------------------------------------------------------
// Wave32 sum via V_WMMA_F32_16X16X4_F32 (codegen-verified in round 1).
// A (16x4 f32): lanes 0-15 hold K=0/K=1, lanes 16-31 hold K=2/K=3. With
// a = {s, 0}: A[M][0] = s_M, A[M][2] = s_{M+16}. B = ones(4x16) =>
// D[M][N] = s_M + s_{M+16}. D: VGPR j, low lanes -> M=j, high lanes -> M=j+8,
// so sum(c[0..7]) on a low lane covers M=0..7 and on a high lane M=8..15.
// Requires EXEC all-ones at call site (call only from converged code).
// ---------------------------------------------------------------------------
__device__ __forceinline__ float wave_sum_wmma(float s) {
    v2f a; a[0] = s;   a[1] = 0.0f;
    v2f b; b[0] = 1.0f; b[1] = 1.0f;
    v8f c = {0.f, 0.f, 0.f, 0.f, 0.f, 0.f, 0.f, 0.f};
    c = __builtin_amdgcn_wmma_f32_16x16x4_f32(
        /*neg_a=*/false, a, /*neg_b=*/false, b,
        /*c_mod=*/(short)0, c, /*reuse_a=*/false, /*reuse_b=*/false);
    float t = ((c[0] + c[1]) + (c[2] + c[3])) + ((c[4] + c[5]) + (c[6] + c[7]));
    return __shfl(t, 0, 32) + __shfl(t, 16, 32);
}

// Focal-loss term for one logit; t in {0,1}; alpha=0.25, gamma=2.
// Single transcendental chain: qe = exp(-|x|) feeds both log1p and sigmoid.
__device__ __forceinline__ float focal_term(float x, float t) {
    const float ax  = fabsf(x);
    const float qe  = expf(-ax);
    const float ce  = fmaxf(x, 0.0f) - x * t + log1pf(qe);
    const float in1 = 1.0f / (1.0f + qe);
    const float p   = (x >= 0.0f) ? in1 : qe * in1;   // sigmoid(x)
    const float pt  = (t > 0.0f) ? p : (1.0f - p);
    const float w   = (t > 0.0f) ? 0.25f : 0.75f;
    const float om  = 1.0f - pt;
    return w * ce * om * om;
}

// ---------------------------------------------------------------------------
// Kernel 0: materialize per-proposal class label once:
//   lab[bq] = matched ? gt_label[b, inds[bq]] : C  (C => all-zero one-hot row)
// ---------------------------------------------------------------------------
__global__ __launch_bounds__(LAB_THREADS)
void label_kernel(const int* __restrict__ gt_label,
                  const int* __restrict__ inds,
                  const int* __restrict__ matched,
                  int*       __restrict__ labs) {
    const int bq = blockIdx.x * LAB_THREADS + threadIdx.x;
    const int b  = bq >> 12;                        // log2(Q_) = 12
    labs[bq] = (matched[bq] != 0) ? gt_label[b * G_ + inds[bq]] : C_;
}

// ---------------------------------------------------------------------------
// Kernel 1: sigmoid focal loss, raw sum. float4 non-temporal stream of the
// 134 MB logits (the roofline-dominant part: ~6 us at 23.3 TB/s); one
// broadcast label dword per 512 B row. Exactly 8 float4 per thread, so the
// trip count is uniform and EXEC is all-ones at the WMMA reduction.
// ---------------------------------------------------------------------------
__global__ __launch_bounds__(FOCAL_THREADS)
void focal_kernel(const float* __restrict__ logits,
                  const int*   __restrict__ labs,
                  float*       __restrict__ partials) {
    const int stride = gridDim.x * blockDim.x;
    float acc = 0.0f;
    for (int e4 = blockIdx.x * blockDim.x + threadIdx.x; e4 < N4_; e4 += stride) {
        const v4f x   = __builtin_nontemporal_load(((const v4f*)logits) + e4);
        const int lab = labs[e4 >> 5];              // 32 float4 per (b,q) row
        const int cb  = (e4 & 31) << 2;
#pragma unroll
        for (int k = 0; k < 4; ++k)
            acc += focal_term(x[k], (cb + k == lab) ? 1.0f : 0.0f);
    }
    const float wsum = wave_sum_wmma(acc);
    __shared__ float red[FOCAL_THREADS / 32];
    if ((threadIdx.x & 31) == 0) red[threadIdx.x >> 5] = wsum;
    __syncthreads();
    if (threadIdx.x == 0) {
        float s = 0.0f;
        for (int i = 0; i < FOCAL_THREADS / 32; ++i) s += red[i];
        partials[blockIdx.x] = s;
    }
}

// Fallback (used only if the workspace is too small for the label array):
// same math, gathers mask/label inline.
__global__ __launch_bounds__(FOCAL_THREADS)
void focal_kernel_fb(const float* __restrict__ logits,
                     const int*   __restrict__ gt_label,
                     const int*   __restrict__ inds,
                     const int*   __restrict__ matched,
                     float*       __restrict__ partials) {
    const int stride = gridDim.x * blockDim.x;
    float acc = 0.0f;
    for (int e4 = blockIdx.x * blockDim.x + threadIdx.x; e4 < N4_; e4 += stride) {
        const v4f x  = __builtin_nontemporal_load(((const v4f*)logits) + e4);
        const int bq = e4 >> 5;
        const int b  = bq >> 12;
        int lab = C_;
        if (matched[bq] != 0) lab = gt_label[b * G_ + inds[bq]];
        const int cb = (e4 & 31) << 2;
#pragma unroll
        for (int k = 0; k < 4; ++k)
            acc += focal_term(x[k], (cb + k == lab) ? 1.0f : 0.0f);
    }
    const float wsum = wave_sum_wmma(acc);
    __shared__ float red[FOCAL_THREADS / 32];
    if ((threadIdx.x & 31) == 0) red[threadIdx.x >> 5] = wsum;
    __syncthreads();
    if (threadIdx.x == 0) {
        float s = 0.0f;
        for (int i = 0; i < FOCAL_THREADS / 32; ++i) s += red[i];
        partials[blockIdx.x] = s;
    }
}

// ---------------------------------------------------------------------------
// Kernel 2: per-proposal L1(center) + L1(size) + (1 - DIoU_rot3d), masked sum.
// One thread per (b,q). Candidate-vertex scratch (<=24 points) kept in LDS.
// ---------------------------------------------------------------------------
__global__ __launch_bounds__(GEOM_THREADS)
void geom_kernel(const float* __restrict__ cpred,
                 const float* __restrict__ spred,
                 const float* __restrict__ apred,
                 const float* __restrict__ gtc,
                 const float* __restrict__ gts,
                 const float* __restrict__ gta,
                 const int*   __restrict__ inds,
                 const int*   __restrict__ matched,
                 float*       __restrict__ partials) {
    __shared__ float s_vx[GEOM_THREADS][24];
    __shared__ float s_vy[GEOM_THREADS][24];
    __shared__ float s_an[GEOM_THREADS][24];
    __shared__ float red[GEOM_THREADS / 32];

    const int tid = threadIdx.x;
    const int bq  = blockIdx.x * GEOM_THREADS + tid;
    const int b   = bq >> 12;                       // log2(Q_) = 12

    float acc = 0.0f;
    if (matched[bq] != 0) {
        const int gi = b * G_ + inds[bq];
        const float c1x = cpred[3 * bq + 0], c1y = cpred[3 * bq + 1], c1z = cpred[3 * bq + 2];
        const float s1x = spred[3 * bq + 0], s1y = spred[3 * bq + 1], s1z = spred[3 * bq + 2];
        const float a1  = apred[bq];
        const float g_cx = gtc[3 * gi + 0], g_cy = gtc[3 * gi + 1], g_cz = gtc[3 * gi + 2];
        const float g_sx = gts[3 * gi + 0], g_sy = gts[3 * gi + 1], g_sz = gts[3 * gi + 2];
        const float a2   = gta[gi];

        // L1 center + size
        acc += fabsf(c1x - g_cx) + fabsf(c1y - g_cy) + fabsf(c1z - g_cz);
        acc += fabsf(s1x - g_sx) + fabsf(s1y - g_sy) + fabsf(s1z - g_sz);

        // ---- box corners (x,y plane) ----
        const float tx[4] = { 0.5f, -0.5f, -0.5f,  0.5f };
        const float ty[4] = { 0.5f,  0.5f, -0.5f, -0.5f };
        float ca1, sa1, ca2, sa2;
        sincosf(a1, &sa1, &ca1);
        sincosf(a2, &sa2, &ca2);
        float p1x[4], p1y[4], p2x[4], p2y[4];
        float xmx = -1e30f, xmn = 1e30f, ymx = -1e30f, ymn = 1e30f;
#pragma unroll
        for (int k = 0; k < 4; ++k) {
            float xx = tx[k] * s1x, yy = ty[k] * s1y;
            p1x[k] = ca1 * xx - sa1 * yy + c1x;
            p1y[k] = sa1 * xx + ca1 * yy + c1y;
            xx = tx[k] * g_sx; yy = ty[k] * g_sy;
            p2x[k] = ca2 * xx - sa2 * yy + g_cx;
            p2y[k] = sa2 * xx + ca2 * yy + g_cy;
            xmx = fmaxf(xmx, fmaxf(p1x[k], p2x[k]));
            xmn = fminf(xmn, fminf(p1x[k], p2x[k]));
            ymx = fmaxf(ymx, fmaxf(p1y[k], p2y[k]));
            ymn = fminf(ymn, fminf(p1y[k], p2y[k]));
        }

        float* vx = &s_vx[tid][0];
        float* vy = &s_vy[tid][0];
        float* an = &s_an[tid][0];
        int m = 0;
        const float e6 = 1e-6f;

        // corners of box1 inside box2 (a=corner0, b=corner1, d=corner3)
        {
            const float abx = p2x[1] - p2x[0], aby = p2y[1] - p2y[0];
            const float adx = p2x[3] - p2x[0], ady = p2y[3] - p2y[0];
            const float nab = abx * abx + aby * aby, nad = adx * adx + ady * ady;
#pragma unroll
            for (int k = 0; k < 4; ++k) {
                const float apx = p1x[k] - p2x[0], apy = p1y[k] - p2y[0];
                const float pab = apx * abx + apy * aby;
                const float pad = apx * adx + apy * ady;
                if (pab > -e6 && pab < nab + e6 && pad > -e6 && pad < nad + e6) {
                    vx[m] = p1x[k]; vy[m] = p1y[k]; ++m;
                }
            }
        }
        // corners of box2 inside box1
        {
            const float abx = p1x[1] - p1x[0], aby = p1y[1] - p1y[0];
            const float adx = p1x[3] - p1x[0], ady = p1y[3] - p1y[0];
            const float nab = abx * abx + aby * aby, nad = adx * adx + ady * ady;
#pragma unroll
            for (int k = 0; k < 4; ++k) {
                const float apx = p2x[k] - p1x[0], apy = p2y[k] - p1y[0];
                const float pab = apx * abx + apy * aby;
                const float pad = apx * adx + apy * ady;
                if (pab > -e6 && pab < nab + e6 && pad > -e6 && pad < nad + e6) {
                    vx[m] = p2x[k]; vy[m] = p2y[k]; ++m;
                }
            }
        }
        // 4x4 edge-edge intersections (same flatten order as reference)
#pragma unroll
        for (int i = 0; i < 4; ++i) {
            const float x1 = p1x[i], y1 = p1y[i];
            const float x2 = p1x[(i + 1) & 3], y2 = p1y[(i + 1) & 3];
#pragma unroll
            for (int j = 0; j < 4; ++j) {
                const float x3 = p2x[j], y3 = p2y[j];
                const float x4 = p2x[(j + 1) & 3], y4 = p2y[(j + 1) & 3];
                const float den = (x2 - x1) * (y4 - y3) - (y2 - y1) * (x4 - x3);
                if (fabsf(den) > 1e-8f) {
                    const float t = ((x3 - x1) * (y4 - y3) - (y3 - y1) * (x4 - x3)) / den;
                    const float u = ((x3 - x1) * (y2 - y1) - (y3 - y1) * (x2 - x1)) / den;
                    if (t > 0.0f && t < 1.0f && u > 0.0f && u < 1.0f) {
                        vx[m] = x1 + t * (x2 - x1);
                        vy[m] = y1 + t * (y2 - y1);
                        ++m;
                    }
                }
            }
        }

        // mean of valid vertices, relative coords, polar angle
        float sx = 0.0f, sy = 0.0f;
        for (int k = 0; k < m; ++k) { sx += vx[k]; sy += vy[k]; }
        const float inv = 1.0f / (float)((m > 1) ? m : 1);
        const float mx = sx * inv, my = sy * inv;
        for (int k = 0; k < m; ++k) {
            vx[k] -= mx; vy[k] -= my;
            an[k] = atan2f(vy[k], vx[k]);
        }
        // stable insertion sort by angle (matches stable argsort of reference)
        for (int i = 1; i < m; ++i) {
            const float ai = an[i], xi = vx[i], yi = vy[i];
            int j = i - 1;
            while (j >= 0 && an[j] > ai) {
                an[j + 1] = an[j]; vx[j + 1] = vx[j]; vy[j + 1] = vy[j]; --j;
            }
            an[j + 1] = ai; vx[j + 1] = xi; vy[j + 1] = yi;
        }
        // shoelace over the m valid sorted vertices (padding contributes 0)
        float cr = 0.0f;
        for (int i = 0; i < m; ++i) {
            const int j = (i + 1 == m) ? 0 : i + 1;
            cr += vx[i] * vy[j] - vy[i] * vx[j];
        }
        const float inter = 0.5f * fabsf(cr);

        // ---- 3D DIoU ----
        const float zmax1 = c1z + 0.5f * s1z, zmin1 = c1z - 0.5f * s1z;
        const float zmax2 = g_cz + 0.5f * g_sz, zmin2 = g_cz - 0.5f * g_sz;
        const float zov = fmaxf(fminf(zmax1, zmax2) - fmaxf(zmin1, zmin2), 0.0f);
        const float inter3d = inter * zov;
        const float vol1 = s1x * s1y * s1z;
        const float vol2 = g_sx * g_sy * g_sz;
        const float uni3d = vol1 + vol2 - inter3d;
        const float zmx = fmaxf(zmax1, zmax2), zmn = fminf(zmin1, zmin2);
        // NOTE: reference's r2 uses box[..., :3] of the *selected* 5-vector,
        // i.e. (center_x, center_y, size_x). Replicate exactly.
        const float dx = c1x - g_cx, dy = c1y - g_cy, dw = s1x - g_sx;
        const float r2 = dx * dx + dy * dy + dw * dw;
        const float ex = xmn - xmx, ey = ymn - ymx, ez = zmn - zmx;
        const float cc2 = ex * ex + ey * ey + ez * ez;
        const float diou = inter3d / uni3d - r2 / cc2;
        acc += 1.0f - diou;
    }

    // converged here: wave sum on matrix pipe, then combine the 2 waves
    const float wsum = wave_sum_wmma(acc);
    if ((tid & 31) == 0) red[tid >> 5] = wsum;
    __syncthreads();
    if (tid == 0) partials[blockIdx.x] = red[0] + red[1];
}

// ---------------------------------------------------------------------------
// Kernel 3: deterministic final reduction and scale by 1/num_boxes.
// ---------------------------------------------------------------------------
__global__ __launch_bounds__(256)
void reduce_kernel(const float* __restrict__ pf, int nf,
                   const float* __restrict__ pg, int ng,
                   const int*   __restrict__ num_boxes,
                   float*       __restrict__ out) {
    __shared__ float red[256];
    float s = 0.0f;
    for (int i = threadIdx.x; i < nf; i += 256) s += pf[i];
    for (int i = threadIdx.x; i < ng; i += 256) s += pg[i];
    red[threadIdx.x] = s;
    __syncthreads();
    for (int k = 128; k > 0; k >>= 1) {
        if (threadIdx.x < k) red[threadIdx.x] += red[threadIdx.x + k];
        __syncthreads();
    }
    if (threadIdx.x == 0) out[0] = red[0] / (float)num_boxes[0];
}

// ---------------------------------------------------------------------------
extern "C" void kernel_launch(void* const* d_in, const int* in_sizes, int n_in,
                              void* d_out, int out_size, void* d_ws, size_t ws_size,
                              hipStream_t stream) {
    const float* logits  = (const float*)d_in[0];
    const float* cpred   = (const float*)d_in[1];
    const float* spred   = (const float*)d_in[2];
    const float* apred   = (const float*)d_in[3];
    const float* gtc     = (const float*)d_in[4];
    const float* gts     = (const float*)d_in[5];
    const float* gta     = (const float*)d_in[6];
    const int*   gtlab   = (const int*)d_in[7];
    const int*   inds    = (const int*)d_in[8];
    const int*   matched = (const int*)d_in[9];
    const int*   nboxes  = (const int*)d_in[10];

    float* p_focal = (float*)d_ws;                 // FOCAL_BLOCKS floats
    float* p_geom  = p_focal + FOCAL_BLOCKS;       // GEOM_BLOCKS floats
    int*   labs    = (int*)(p_geom + GEOM_BLOCKS); // BQ_ ints
    const size_t need = (size_t)(FOCAL_BLOCKS + GEOM_BLOCKS) * sizeof(float)
                      + (size_t)BQ_ * sizeof(int);

    if (ws_size >= need) {
        label_kernel<<<LAB_BLOCKS, LAB_THREADS, 0, stream>>>(gtlab, inds, matched, labs);
        focal_kernel<<<FOCAL_BLOCKS, FOCAL_THREADS, 0, stream>>>(logits, labs, p_focal);
    } else {
        focal_kernel_fb<<<FOCAL_BLOCKS, FOCAL_THREADS, 0, stream>>>(
            logits, gtlab, inds, matched, p_focal);
    }
    geom_kernel<<<GEOM_BLOCKS, GEOM_THREADS, 0, stream>>>(
        cpred, spred, apred, gtc, gts, gta, inds, matched, p_geom);
    reduce_kernel<<<1, 256, 0, stream>>>(
        p_focal, FOCAL_BLOCKS, p_geom, GEOM_BLOCKS, nboxes, (float*)d_out);
}